// ScaledDotProductAttention_49409303773924
// MI455X (gfx1250) — compile-verified
//
#include <hip/hip_runtime.h>

typedef float v2f __attribute__((ext_vector_type(2)));
typedef float v4f __attribute__((ext_vector_type(4)));
typedef float v8f __attribute__((ext_vector_type(8)));

#define BH 64
#define L 128
#define D 128
static constexpr float SCALE = 0.08838834764831845f; // 1/sqrt(128)

// ---------------------------------------------------------------------------
// Kernel 1: v2[b,l,m] = v[b,l,m] + sum_d v[b,l,d] * relv[b,l,m,d]
// One wave per (b,l) row. relv streamed once -> non-temporal loads.
// ---------------------------------------------------------------------------
__global__ __launch_bounds__(256) void v2_kernel(const float* __restrict__ v,
                                                 const float* __restrict__ relv,
                                                 float* __restrict__ v2out) {
  const int wave = threadIdx.x >> 5;
  const int lane = threadIdx.x & 31;
  const int row  = blockIdx.x * 8 + wave;            // row = b*L + l
  __shared__ float vs[8][D];
  const v4f* vrow = (const v4f*)(v + (size_t)row * D);
  ((v4f*)vs[wave])[lane] = vrow[lane];
  __syncthreads();

  const float* R = relv + (size_t)row * L * D;       // 128 x 128 tile
  #pragma unroll
  for (int j = 0; j < 4; ++j) {
    const int m = lane + 32 * j;
    const v4f* Rm = (const v4f*)(R + (size_t)m * D);
    float s = 0.f;
    #pragma unroll 8
    for (int c = 0; c < 32; ++c) {
      v4f rv = __builtin_nontemporal_load(&Rm[c]);
      v4f vv = ((const v4f*)vs[wave])[c];
      s += rv.x * vv.x + rv.y * vv.y + rv.z * vv.z + rv.w * vv.w;
    }
    v2out[(size_t)row * D + m] = vs[wave][m] + s;
  }
}

// ---------------------------------------------------------------------------
// Kernel 2: scores = QK^T (WMMA f32 16x16x4, one m-tile per wave)
//           + q . relq (streamed by all 8 waves), then softmax.
// One 256-thread block per (b, 16-row l-tile): 8x the waves on the
// latency-bound relq stream vs. the single-wave version.
// ---------------------------------------------------------------------------
__global__ __launch_bounds__(256) void attn_kernel(const float* __restrict__ q,
                                                   const float* __restrict__ k,
                                                   const float* __restrict__ relq,
                                                   float* __restrict__ attn_out) {
  const int b    = blockIdx.x >> 3;
  const int lt   = blockIdx.x & 7;
  const int tid  = threadIdx.x;
  const int wave = tid >> 5;
  const int lane = tid & 31;
  const int half = lane >> 4;     // 0: lanes 0-15, 1: lanes 16-31
  const int ln   = lane & 15;

  __shared__ float qs[16][D];
  __shared__ float sc[16][L];

  // stage 16x128 Q tile in LDS (2048 floats, 2 v4f per thread)
  const v4f* qg  = (const v4f*)(q + ((size_t)b * L + (size_t)lt * 16) * D);
  v4f*       qls = (v4f*)&qs[0][0];
  #pragma unroll
  for (int i = 0; i < 2; ++i) qls[tid + 256 * i] = qg[tid + 256 * i];
  __syncthreads();

  // ---- QK^T via v_wmma_f32_16x16x4_f32; wave w owns m-tile w ----
  {
    const int mt = wave;
    v8f acc = {};
    const float* km = k + (size_t)b * L * D + (size_t)(mt * 16 + ln) * D;
    #pragma unroll 4
    for (int kc = 0; kc < 32; ++kc) {
      const int kk = kc * 4 + 2 * half;
      v2f a;  a.x  = qs[ln][kk];  a.y  = qs[ln][kk + 1]; // A[M=ln][K=kk,kk+1]
      v2f bb; bb.x = km[kk];      bb.y = km[kk + 1];     // B[K][N=ln]
      acc = __builtin_amdgcn_wmma_f32_16x16x4_f32(false, a, false, bb,
                                                  (short)0, acc, false, false);
    }
    #pragma unroll
    for (int r = 0; r < 8; ++r) sc[r + 8 * half][mt * 16 + ln] = acc[r];
  }
  __syncthreads();

  // ---- relative-q contribution: 2048 (l,m) dots, 8 per thread ----
  const float* Rq = relq + ((size_t)b * L + (size_t)lt * 16) * (size_t)L * D;
  #pragma unroll
  for (int i = 0; i < 8; ++i) {
    const int p = tid + 256 * i;
    const int l = p >> 7;
    const int m = p & 127;
    const v4f* Rr  = (const v4f*)(Rq + ((size_t)l * L + m) * D);
    const v4f* qv4 = (const v4f*)qs[l];
    float s = 0.f;
    #pragma unroll 8
    for (int c = 0; c < 32; ++c) {
      v4f rv = __builtin_nontemporal_load(&Rr[c]);
      v4f qv = qv4[c];
      s += rv.x * qv.x + rv.y * qv.y + rv.z * qv.z + rv.w * qv.w;
    }
    sc[l][m] += s;
  }
  __syncthreads();

  // ---- softmax over m: wave w handles rows 2w, 2w+1 ----
  float* arow = attn_out + ((size_t)b * L + (size_t)lt * 16) * L;
  #pragma unroll
  for (int li = 0; li < 2; ++li) {
    const int l = wave * 2 + li;
    float x[4];
    #pragma unroll
    for (int j = 0; j < 4; ++j) x[j] = sc[l][lane + 32 * j] * SCALE;
    float mx = fmaxf(fmaxf(x[0], x[1]), fmaxf(x[2], x[3]));
    #pragma unroll
    for (int off = 16; off > 0; off >>= 1) mx = fmaxf(mx, __shfl_xor(mx, off, 32));
    float sum = 0.f;
    #pragma unroll
    for (int j = 0; j < 4; ++j) { x[j] = __expf(x[j] - mx); sum += x[j]; }
    #pragma unroll
    for (int off = 16; off > 0; off >>= 1) sum += __shfl_xor(sum, off, 32);
    const float inv = 1.0f / sum;
    #pragma unroll
    for (int j = 0; j < 4; ++j) arow[(size_t)l * L + lane + 32 * j] = x[j] * inv;
  }
}

// ---------------------------------------------------------------------------
// Kernel 3: context[b] = attn[b] (128x128) @ v2[b] (128x128) via WMMA.
// One block per b; v2[b] staged in 64KB LDS, then overwritten in-place.
// ---------------------------------------------------------------------------
__global__ __launch_bounds__(256) void ctx_kernel(const float* __restrict__ attn,
                                                  const float* __restrict__ v2in,
                                                  float* __restrict__ ctx) {
  const int b = blockIdx.x;
  __shared__ float v2s[L][D];                         // 64 KB
  const v4f* src = (const v4f*)(v2in + (size_t)b * L * D);
  v4f*       dst = (v4f*)&v2s[0][0];
  #pragma unroll
  for (int i = 0; i < 16; ++i) dst[threadIdx.x + 256 * i] = src[threadIdx.x + 256 * i];
  __syncthreads();                                    // all v2 reads done before ctx writes

  const int wave = threadIdx.x >> 5;
  const int lane = threadIdx.x & 31;
  const int half = lane >> 4;
  const int ln   = lane & 15;
  const int l0   = wave * 16;
  const float* A = attn + ((size_t)b * L + l0) * L;   // 16 x 128 attn tile

  v8f zero = {};
  v8f acc[8];
  #pragma unroll
  for (int dt = 0; dt < 8; ++dt) acc[dt] = zero;

  for (int kc = 0; kc < 32; ++kc) {
    const int kk = kc * 4 + 2 * half;
    v2f a; a.x = A[(size_t)ln * L + kk]; a.y = A[(size_t)ln * L + kk + 1];
    #pragma unroll
    for (int dt = 0; dt < 8; ++dt) {
      v2f bb; bb.x = v2s[kk][dt * 16 + ln]; bb.y = v2s[kk + 1][dt * 16 + ln];
      acc[dt] = __builtin_amdgcn_wmma_f32_16x16x4_f32(false, a, false, bb,
                                                      (short)0, acc[dt], false, false);
    }
  }
  #pragma unroll
  for (int dt = 0; dt < 8; ++dt) {
    #pragma unroll
    for (int r = 0; r < 8; ++r) {
      ctx[((size_t)b * L + l0 + r + 8 * half) * D + dt * 16 + ln] = acc[dt][r];
    }
  }
}

// ---------------------------------------------------------------------------
extern "C" void kernel_launch(void* const* d_in, const int* in_sizes, int n_in,
                              void* d_out, int out_size, void* d_ws, size_t ws_size,
                              hipStream_t stream) {
  (void)in_sizes; (void)n_in; (void)out_size; (void)d_ws; (void)ws_size;
  const float* q    = (const float*)d_in[0];
  const float* k    = (const float*)d_in[1];
  const float* v    = (const float*)d_in[2];
  const float* relq = (const float*)d_in[3];
  const float* relv = (const float*)d_in[4];

  float* ctx  = (float*)d_out;                       // context: BH*L*D floats
  float* attn = ctx + (size_t)BH * L * D;            // attn:    BH*L*L floats

  // v2 is staged in the context region (fully overwritten by ctx_kernel).
  v2_kernel  <<<BH * L / 8,    256, 0, stream>>>(v, relv, ctx);
  attn_kernel<<<BH * (L / 16), 256, 0, stream>>>(q, k, relq, attn);
  ctx_kernel <<<BH,            256, 0, stream>>>(attn, ctx, ctx);
}